// SSSM_55688545960276
// MI455X (gfx1250) — compile-verified
//
#include <hip/hip_runtime.h>
#include <math.h>

// ---------------- problem constants ----------------
#define B_    8
#define CH_   19
#define PN_   128
#define PS_   200
#define L_    2432      // CH_*PN_
#define LP_   (L_ + 2)  // padded pitch for conv3 input (zero halo both sides)
#define RES_  256
#define H2_   512
#define NB_   6
#define S4C_  4
#define NF_   101       // rfft bins for length-200
#define FFTN  8192      // power-of-2 >= 2L-1 : same linear convolution as ref's n=4864 circular
#define FFTLOG 13

typedef __attribute__((ext_vector_type(2))) float v2f;
typedef __attribute__((ext_vector_type(8))) float v8f;

__device__ __forceinline__ float gelu_f(float x) {
  return 0.5f * x * (1.0f + erff(x * 0.70710678118654752f));   // exact GELU
}
__device__ __forceinline__ float sigm_f(float x) { return 1.0f / (1.0f + expf(-x)); }

// fp32 WMMA: D(16x16,f32) = A(16x4,f32) * B(4x16,f32) + C
__device__ __forceinline__ v8f wmma4(v2f a, v2f b, v8f c) {
  return __builtin_amdgcn_wmma_f32_16x16x4_f32(false, a, false, b, (short)0, c, false, false);
}

// ---- CDNA5 async global->LDS copy (ASYNCcnt-tracked) ----
__device__ __forceinline__ void async_g2l_b32(void* lds_dst, const float* gsrc) {
  unsigned lo = (unsigned)(size_t)lds_dst;                 // low 32 bits = LDS byte offset
  unsigned long long ga = (unsigned long long)(size_t)gsrc;
  asm volatile("global_load_async_to_lds_b32 %0, %1, off" :: "v"(lo), "v"(ga) : "memory");
}
__device__ __forceinline__ void wait_async0() {
  asm volatile("s_wait_asynccnt 0" ::: "memory");
}

// =====================================================================
// Generic batched WMMA GEMM:  C[b](MxN) = act(alpha*A(MxK)*X[b](KxN)+bias) [+Cold if ACC]
// Block = 128 threads (4 waves); each wave owns a 16x64 C strip (4 WMMA tiles, A reused).
// OOB rows/cols handled by ADDRESS CLAMPING (results discarded at store) -> branch-free body.
// KGUARD only for K % 4 != 0 (zeroes the A fragment; 0*garbage = 0).
// grid = (ceil(N/256), ceil(M/16), B)
// =====================================================================
template<int ACT, int ACC, int KGUARD>   // ACT: 0=none 1=relu 2=gelu
__global__ void k_gemm(const float* __restrict__ A, const float* __restrict__ X,
                       const float* __restrict__ bias, float* __restrict__ C,
                       int M, int K, int N, float alpha)
{
  const int lane = threadIdx.x & 31;
  const int wv   = threadIdx.x >> 5;
  const int mt   = blockIdx.y << 4;
  const int nt0  = (blockIdx.x << 8) + (wv << 6);
  const int b    = blockIdx.z;
  const float* Xb = X + (size_t)b * K * N;
  float* Cb = C + (size_t)b * M * N;
  const int l16  = lane & 15;
  const int koff = (lane >> 4) << 1;       // 0 or 2
  const int arow = (mt + l16 < M) ? (mt + l16) : (M - 1);   // clamp (garbage rows never stored)
  int nc[4];
#pragma unroll
  for (int j = 0; j < 4; ++j) {
    const int n = nt0 + (j << 4) + l16;
    nc[j] = (n < N) ? n : (N - 1);                          // clamp (garbage cols never stored)
  }
  v8f zero = {};
  v8f acc[4] = {zero, zero, zero, zero};
  for (int k0 = 0; k0 < K; k0 += 4) {
    const int ka  = k0 + koff;
    const int kc0 = KGUARD ? ((ka     < K) ? ka     : (K - 1)) : ka;
    const int kc1 = KGUARD ? ((ka + 1 < K) ? ka + 1 : (K - 1)) : ka + 1;
    v2f a;
    a.x = A[(size_t)arow * K + kc0];
    a.y = A[(size_t)arow * K + kc1];
    if (KGUARD) {                 // zeroing A alone kills OOB-K contributions
      if (ka     >= K) a.x = 0.0f;
      if (ka + 1 >= K) a.y = 0.0f;
    }
    if (ka + 64 < K) __builtin_prefetch(&A[(size_t)arow * K + ka + 64], 0, 0);
#pragma unroll
    for (int j = 0; j < 4; ++j) {
      v2f bb;
      bb.x = Xb[(size_t)kc0 * N + nc[j]];
      bb.y = Xb[(size_t)kc1 * N + nc[j]];
      acc[j] = wmma4(a, bb, acc[j]);
    }
  }
#pragma unroll
  for (int j = 0; j < 4; ++j) {
    const int n = nt0 + (j << 4) + l16;
    if (n >= N) continue;
#pragma unroll
    for (int r = 0; r < 8; ++r) {
      const int mrow = mt + r + ((lane >> 4) << 3);
      if (mrow >= M) continue;
      float val = alpha * acc[j][r];
      if (bias) val += bias[mrow];
      const size_t ci = (size_t)mrow * N + n;
      if (ACC) val += Cb[ci];
      if (ACT == 1)      val = fmaxf(val, 0.0f);
      else if (ACT == 2) val = gelu_f(val);
      Cb[ci] = val;
    }
  }
}

// =====================================================================
// conv3 (k=3, pad=1, 256->512) as implicit-im2col WMMA GEMM over K=768.
// Weights pre-repacked t-major:  WC[o, t*256+c]  (so the K index is (t,c) via shift/mask,
// and a 4-aligned k-block never crosses a t boundary -> contiguous A loads).
// Input pre-summed & zero-padded:  XP[b, c, 1+l]  pitch LP_ -> branch-free B loads.
// grid = (ceil(L/256), 32, B), block 128
// =====================================================================
__global__ void k_conv3(const float* __restrict__ WC, const float* __restrict__ XP,
                        const float* __restrict__ bias, float* __restrict__ U)
{
  const int Kc   = 3 * RES_;                 // 768
  const int lane = threadIdx.x & 31;
  const int wv   = threadIdx.x >> 5;
  const int mt   = blockIdx.y << 4;
  const int nt0  = (blockIdx.x << 8) + (wv << 6);
  const int b    = blockIdx.z;
  const float* Xb = XP + (size_t)b * RES_ * LP_;
  float* Ub = U + (size_t)b * H2_ * L_;
  const int l16  = lane & 15;
  const int koff = (lane >> 4) << 1;
  const float* Arow = WC + (size_t)(mt + l16) * Kc;
  int nc[4];
#pragma unroll
  for (int j = 0; j < 4; ++j) {
    const int n = nt0 + (j << 4) + l16;
    nc[j] = (n < L_) ? n : (L_ - 1);         // clamp; garbage cols never stored
  }
  v8f zero = {};
  v8f acc[4] = {zero, zero, zero, zero};
  for (int k0 = 0; k0 < Kc; k0 += 4) {
    const int ka = k0 + koff;
    const int t  = ka >> 8;                  // tap 0..2 (same for ka and ka+1: ka is even)
    const int c  = ka & 255;
    v2f a;
    a.x = Arow[ka];
    a.y = Arow[ka + 1];
#pragma unroll
    for (int j = 0; j < 4; ++j) {
      const size_t ix = (size_t)c * LP_ + nc[j] + t;   // lp = l + t (halo handles pad)
      v2f bb;
      bb.x = Xb[ix];
      bb.y = Xb[ix + LP_];                   // channel c+1, same column
      acc[j] = wmma4(a, bb, acc[j]);
    }
  }
#pragma unroll
  for (int j = 0; j < 4; ++j) {
    const int n = nt0 + (j << 4) + l16;
    if (n >= L_) continue;
#pragma unroll
    for (int r = 0; r < 8; ++r) {
      const int mrow = mt + r + ((lane >> 4) << 3);
      Ub[(size_t)mrow * L_ + n] = gelu_f(acc[j][r] + bias[mrow]);
    }
  }
}

// weight repack: WC[o, t*256 + c] = cw[(o*256 + c)*3 + t]
__global__ void k_repack_cw(const float* __restrict__ cw, float* __restrict__ WC) {
  const int n = H2_ * 3 * RES_;
  for (int i = blockIdx.x * blockDim.x + threadIdx.x; i < n; i += gridDim.x * blockDim.x) {
    const int o = i / (3 * RES_);
    const int r = i - o * (3 * RES_);
    const int t = r >> 8, c = r & 255;
    WC[i] = cw[((size_t)o * RES_ + c) * 3 + t];
  }
}

// padded input build: XP[b,c,0]=XP[b,c,L+1]=0 ; XP[b,c,1+l] = hx+noise
__global__ void k_xin(const float* __restrict__ hx, const float* __restrict__ no,
                      float* __restrict__ xp) {
  const size_t n = (size_t)B_ * RES_ * LP_;
  for (size_t i = (size_t)blockIdx.x * blockDim.x + threadIdx.x; i < n;
       i += (size_t)gridDim.x * blockDim.x) {
    const size_t row = i / LP_;
    const int lp = (int)(i - row * LP_);
    float v = 0.0f;
    if (lp >= 1 && lp <= L_) v = hx[row * L_ + (lp - 1)] + no[row * L_ + (lp - 1)];
    xp[i] = v;
  }
}

// =====================================================================
// Radix-2 Stockham FFT in LDS (self-sorting). sign=-1 fwd, +1 inv (unnormalized).
// =====================================================================
__device__ float2* fft_stockham(float2* src, float2* dst, int tid, int nt, float sign) {
  int l = FFTN >> 1;
  int mlog = 0;
  for (int s = 0; s < FFTLOG; ++s) {
    const int m = 1 << mlog;
    for (int idx = tid; idx < (FFTN >> 1); idx += nt) {
      const int j = idx >> mlog;
      float2 c0 = src[idx];
      float2 c1 = src[idx + (FFTN >> 1)];
      const float sx = c0.x + c1.x, sy = c0.y + c1.y;
      const float dx = c0.x - c1.x, dy = c0.y - c1.y;
      const float ang = sign * 3.14159265358979f * (float)j / (float)l;
      float sn, cs; __sincosf(ang, &sn, &cs);
      const int w = idx + j * m;             // k + 2*j*m
      dst[w]     = make_float2(sx, sy);
      dst[w + m] = make_float2(cs * dx - sn * dy, cs * dy + sn * dx);
    }
    __syncthreads();
    float2* t = src; src = dst; dst = t;
    l >>= 1; ++mlog;
  }
  return src;                                // result lives here after final swap
}

// async row load into complex LDS buffer: reals via ASYNC DMA, imag+tail via DS stores
// (disjoint byte ranges -> no ordering hazard between the two paths)
__device__ __forceinline__ void load_row_async(float2* W0, const float* row, int tid, int nt) {
  for (int i = tid; i < FFTN; i += nt) {
    if (i < L_) {
      W0[i].y = 0.0f;                        // DS store to bytes [8i+4, 8i+8)
      async_g2l_b32(&W0[i], row + i);        // ASYNC DMA into bytes [8i, 8i+4)
    } else {
      W0[i] = make_float2(0.0f, 0.0f);
    }
  }
  wait_async0();
  __syncthreads();
}

// kernel-spectrum precompute for one residual block: kf[s,h,0..4096]
__global__ void k_kf(const float* __restrict__ kblk, float2* __restrict__ kf) {
  extern __shared__ float2 sm2[];
  float2* W0 = sm2; float2* W1 = sm2 + FFTN;
  const int h = blockIdx.x, s = blockIdx.y;
  const int tid = threadIdx.x, nt = blockDim.x;
  load_row_async(W0, kblk + ((size_t)s * H2_ + h) * L_, tid, nt);
  float2* R = fft_stockham(W0, W1, tid, nt, -1.0f);
  float2* out = kf + ((size_t)s * H2_ + h) * (FFTN / 2 + 1);
  for (int i = tid; i <= FFTN / 2; i += nt) out[i] = R[i];
}

// fused S4 long conv: per (b,h) 1 fwd FFT + 4x (spectral mult + inv FFT) + u*D skip.
// LDS: 3 x 8192 complex = 192KB (needs CDNA5's 320KB/WGP).
__global__ void k_s4(const float* __restrict__ u, const float2* __restrict__ kf,
                     const float* __restrict__ D, float* __restrict__ y)
{
  extern __shared__ float2 sm3[];
  float2* W0 = sm3; float2* W1 = sm3 + FFTN; float2* UF = sm3 + 2 * FFTN;
  const int h = blockIdx.x, b = blockIdx.y;
  const int tid = threadIdx.x, nt = blockDim.x;
  const float* ur = u + ((size_t)b * H2_ + h) * L_;
  load_row_async(W0, ur, tid, nt);
  float2* R = fft_stockham(W0, W1, tid, nt, -1.0f);
  for (int i = tid; i < FFTN; i += nt) UF[i] = R[i];
  __syncthreads();
  for (int s = 0; s < S4C_; ++s) {
    const float2* kr = kf + ((size_t)s * H2_ + h) * (FFTN / 2 + 1);
    for (int i = tid; i < FFTN; i += nt) {
      float2 kv;
      if (i <= FFTN / 2) kv = kr[i];
      else { float2 t2 = kr[FFTN - i]; kv = make_float2(t2.x, -t2.y); }  // conj symmetry
      const float2 uv = UF[i];
      W0[i] = make_float2(uv.x * kv.x - uv.y * kv.y, uv.x * kv.y + uv.y * kv.x);
    }
    __syncthreads();
    float2* Ri = fft_stockham(W0, W1, tid, nt, 1.0f);
    const float Ds = D[s * H2_ + h];
    float* yr = y + ((size_t)b * (S4C_ * H2_) + s * H2_ + h) * L_;
    for (int l = tid; l < L_; l += nt)
      yr[l] = Ri[l].x * (1.0f / (float)FFTN) + ur[l] * Ds;
    __syncthreads();
  }
}

// =====================================================================
// Patch-embedding kernels
// =====================================================================
__global__ void k_conv0(const float* __restrict__ x, const float* __restrict__ w,
                        const float* __restrict__ bias, float* __restrict__ p0)
{
  __shared__ float xr[PS_];
  __shared__ float wsh[25 * 49];
  const int l = blockIdx.x, b = blockIdx.y, tid = threadIdx.x;
  for (int i = tid; i < PS_; i += blockDim.x) xr[i] = x[((size_t)b * L_ + l) * PS_ + i];
  for (int i = tid; i < 25 * 49; i += blockDim.x) wsh[i] = w[i];
  __syncthreads();
  if (tid < 200) {
    const int o = tid >> 3, ww = tid & 7;
    float acc = bias[o];
    const int base = ww * 25 - 24;
    for (int t = 0; t < 49; ++t) {
      const int ix = base + t;
      if (ix >= 0 && ix < PS_) acc += xr[ix] * wsh[o * 49 + t];
    }
    p0[(((size_t)b * 25 + o) * L_ + l) * 8 + ww] = acc;
  }
}

__global__ void k_gn_gelu(float* __restrict__ p, const float* __restrict__ gw,
                          const float* __restrict__ gb)
{
  __shared__ float s1[256], s2[256];
  __shared__ float mean_s, inv_s;
  const int b = blockIdx.x / 5, g = blockIdx.x % 5;
  const int tid = threadIdx.x;
  const int CE = L_ * 8;
  const int GE = 5 * CE;
  const size_t base = ((size_t)b * 25 + g * 5) * CE;
  float su = 0.0f, sq = 0.0f;
  for (int i = tid; i < GE; i += blockDim.x) { float v = p[base + i]; su += v; sq += v * v; }
  s1[tid] = su; s2[tid] = sq; __syncthreads();
  for (int s = 128; s > 0; s >>= 1) {
    if (tid < s) { s1[tid] += s1[tid + s]; s2[tid] += s2[tid + s]; }
    __syncthreads();
  }
  if (tid == 0) {
    const float m = s1[0] / (float)GE;
    const float v = s2[0] / (float)GE - m * m;
    mean_s = m; inv_s = rsqrtf(v + 1e-5f);
  }
  __syncthreads();
  for (int i = tid; i < GE; i += blockDim.x) {
    const int c = g * 5 + i / CE;
    float v = (p[base + i] - mean_s) * inv_s * gw[c] + gb[c];
    p[base + i] = gelu_f(v);
  }
}

__global__ void k_convw3(const float* __restrict__ in, const float* __restrict__ w,
                         const float* __restrict__ bias, float* __restrict__ out)
{
  __shared__ float tile[200];
  __shared__ float wsh[25 * 25 * 3];
  const int l = blockIdx.x, b = blockIdx.y, tid = threadIdx.x;
  for (int i = tid; i < 200; i += blockDim.x) {
    const int c = i >> 3, ww = i & 7;
    tile[i] = in[(((size_t)b * 25 + c) * L_ + l) * 8 + ww];
  }
  for (int i = tid; i < 25 * 25 * 3; i += blockDim.x) wsh[i] = w[i];
  __syncthreads();
  if (tid < 200) {
    const int o = tid >> 3, ww = tid & 7;
    float acc = bias[o];
    for (int c = 0; c < 25; ++c)
      for (int t = 0; t < 3; ++t) {
        const int wi = ww + t - 1;
        if (wi >= 0 && wi < 8) acc += tile[c * 8 + wi] * wsh[(o * 25 + c) * 3 + t];
      }
    out[(((size_t)b * 25 + o) * L_ + l) * 8 + ww] = acc;
  }
}

__global__ void k_assemble(const float* __restrict__ p2, float* __restrict__ h) {
  const size_t n = (size_t)B_ * PS_ * L_;
  for (size_t i = (size_t)blockIdx.x * blockDim.x + threadIdx.x; i < n;
       i += (size_t)gridDim.x * blockDim.x) {
    const size_t b = i / ((size_t)PS_ * L_);
    const size_t r = i - b * ((size_t)PS_ * L_);
    const int p = (int)(r / L_), l = (int)(r - (size_t)p * L_);
    h[i] = p2[((b * 25 + (p >> 3)) * L_ + l) * 8 + (p & 7)];
  }
}

__global__ void k_tables(float* __restrict__ ct, float* __restrict__ st) {
  const int i = blockIdx.x * blockDim.x + threadIdx.x;
  if (i >= NF_ * PS_) return;
  const int f = i / PS_, q = i - f * PS_;
  const int r = (f * q) % PS_;                      // exact integer phase reduction
  const float ang = 6.28318530717958648f * (float)r / 200.0f;
  ct[i] = cosf(ang); st[i] = sinf(ang);
}

__global__ void k_spec(const float* __restrict__ x, const float* __restrict__ ct,
                       const float* __restrict__ st, float* __restrict__ sp)
{
  __shared__ float xr[PS_];
  const int l = blockIdx.x, b = blockIdx.y, tid = threadIdx.x;
  for (int i = tid; i < PS_; i += blockDim.x) xr[i] = x[((size_t)b * L_ + l) * PS_ + i];
  __syncthreads();
  if (tid < NF_) {
    float re = 0.0f, im = 0.0f;
    for (int q = 0; q < PS_; ++q) { const float v = xr[q]; re += v * ct[tid * PS_ + q]; im += v * st[tid * PS_ + q]; }
    sp[((size_t)b * NF_ + tid) * L_ + l] = sqrtf(re * re + im * im) * (1.0f / 200.0f);
  }
}

__global__ void k_posconv(const float* __restrict__ h, const float* __restrict__ pw,
                          const float* __restrict__ pb, float* __restrict__ h2)
{
  __shared__ float plane[L_];
  __shared__ float wloc[19 * 7];
  const int p = blockIdx.x, b = blockIdx.y, tid = threadIdx.x;
  const size_t row = ((size_t)b * PS_ + p) * L_;
  for (int i = tid; i < L_; i += blockDim.x) plane[i] = h[row + i];
  for (int i = tid; i < 133; i += blockDim.x) wloc[i] = pw[(size_t)p * 133 + i];
  __syncthreads();
  for (int l = tid; l < L_; l += blockDim.x) {
    const int c = l >> 7, n = l & 127;
    float acc = pb[p];
    for (int i = 0; i < 19; ++i) {
      const int ci = c + i - 9;
      if (ci < 0 || ci >= CH_) continue;
      for (int j = 0; j < 7; ++j) {
        const int nj = n + j - 3;
        if (nj < 0 || nj >= PN_) continue;
        acc += plane[ci * 128 + nj] * wloc[i * 7 + j];
      }
    }
    h2[row + l] = plane[l] + acc;
  }
}

// =====================================================================
// small elementwise / reduction kernels
// =====================================================================
__global__ void k_copy(const float* __restrict__ a, float* __restrict__ b, size_t n) {
  for (size_t i = (size_t)blockIdx.x * blockDim.x + threadIdx.x; i < n;
       i += (size_t)gridDim.x * blockDim.x) b[i] = a[i];
}
__global__ void k_zero(float* __restrict__ a, size_t n) {
  for (size_t i = (size_t)blockIdx.x * blockDim.x + threadIdx.x; i < n;
       i += (size_t)gridDim.x * blockDim.x) a[i] = 0.0f;
}

__global__ void k_rms(const float* __restrict__ hx, const float* __restrict__ sn,
                      float* __restrict__ hn)
{
  const int l = blockIdx.x * blockDim.x + threadIdx.x;
  const int b = blockIdx.y;
  if (l >= L_) return;
  const float* base = hx + (size_t)b * RES_ * L_ + l;
  float ss = 0.0f;
  for (int c = 0; c < RES_; ++c) { const float v = base[(size_t)c * L_]; ss += v * v; }
  const float rms = sqrtf(ss) * (1.0f / 16.0f);    // /sqrt(256)
  const float inv = 1.0f / (rms + 1e-8f);
  float* ob = hn + (size_t)b * RES_ * L_ + l;
  for (int c = 0; c < RES_; ++c) ob[(size_t)c * L_] = sn[c] * base[(size_t)c * L_] * inv;
}

__global__ void k_glu(const float* __restrict__ y2, float* __restrict__ ht) {
  const size_t n = (size_t)B_ * H2_ * L_;
  for (size_t i = (size_t)blockIdx.x * blockDim.x + threadIdx.x; i < n;
       i += (size_t)gridDim.x * blockDim.x) {
    const size_t b = i / ((size_t)H2_ * L_);
    const size_t r = i - b * ((size_t)H2_ * L_);
    const size_t o1 = (b * 2 * H2_) * L_ + r;
    const size_t o2 = o1 + (size_t)H2_ * L_;
    ht[i] = y2[o1] * sigm_f(y2[o2]);
  }
}

__global__ void k_gate(const float* __restrict__ ht, const float* __restrict__ ab,
                       float* __restrict__ out) {
  const size_t n = (size_t)B_ * RES_ * L_;
  for (size_t i = (size_t)blockIdx.x * blockDim.x + threadIdx.x; i < n;
       i += (size_t)gridDim.x * blockDim.x) {
    const size_t b = i / ((size_t)RES_ * L_);
    const size_t r = i - b * ((size_t)RES_ * L_);
    const size_t o1 = (b * H2_) * L_ + r;
    const size_t o2 = o1 + (size_t)RES_ * L_;
    const float g1 = ht[o1] + ab[o1];
    const float g2 = ht[o2] + ab[o2];
    out[i] = tanhf(g1) * sigm_f(g2);
  }
}

__global__ void k_combine(const float* __restrict__ hn, const float* __restrict__ tmp,
                          float* __restrict__ hx) {
  const size_t n = (size_t)B_ * RES_ * L_;
  for (size_t i = (size_t)blockIdx.x * blockDim.x + threadIdx.x; i < n;
       i += (size_t)gridDim.x * blockDim.x)
    hx[i] = (hn[i] + tmp[i]) * 0.70710678118654752f;
}

__global__ void k_final_ln(const float* __restrict__ o2, const float* __restrict__ lw,
                           const float* __restrict__ lb, float* __restrict__ out)
{
  const int l = blockIdx.x * blockDim.x + threadIdx.x;
  const int b = blockIdx.y;
  if (l >= L_) return;
  const float* base = o2 + (size_t)b * PS_ * L_ + l;
  float m = 0.0f;
  for (int p = 0; p < PS_; ++p) m += base[(size_t)p * L_];
  m *= (1.0f / 200.0f);
  float va = 0.0f;
  for (int p = 0; p < PS_; ++p) { const float d = base[(size_t)p * L_] - m; va += d * d; }
  va *= (1.0f / 200.0f);
  const float inv = rsqrtf(va + 1e-5f);
  float* ob = out + ((size_t)b * L_ + l) * PS_;
  for (int p = 0; p < PS_; ++p)
    ob[p] = (base[(size_t)p * L_] - m) * inv * lw[p] + lb[p];
}

// =====================================================================
// host orchestration
// =====================================================================
static inline dim3 gemm_grid(int M) { return dim3((L_ + 255) / 256, (M + 15) / 16, B_); }

extern "C" void kernel_launch(void* const* d_in, const int* in_sizes, int n_in,
                              void* d_out, int out_size, void* d_ws, size_t ws_size,
                              hipStream_t stream)
{
  (void)in_sizes; (void)n_in; (void)out_size; (void)ws_size;
  const float* x      = (const float*)d_in[0];
  const float* pe_w   = (const float*)d_in[1];
  const float* pe_b   = (const float*)d_in[2];
  const float* pi0_w  = (const float*)d_in[3];
  const float* pi0_b  = (const float*)d_in[4];
  const float* gn0_w  = (const float*)d_in[5];
  const float* gn0_b  = (const float*)d_in[6];
  const float* pi1_w  = (const float*)d_in[7];
  const float* pi1_b  = (const float*)d_in[8];
  const float* gn1_w  = (const float*)d_in[9];
  const float* gn1_b  = (const float*)d_in[10];
  const float* pi2_w  = (const float*)d_in[11];
  const float* pi2_b  = (const float*)d_in[12];
  const float* gn2_w  = (const float*)d_in[13];
  const float* gn2_b  = (const float*)d_in[14];
  const float* spec_w = (const float*)d_in[15];
  const float* spec_b = (const float*)d_in[16];
  const float* ic_w   = (const float*)d_in[17];
  const float* ic_b   = (const float*)d_in[18];
  const float* blk_sn = (const float*)d_in[19];
  const float* blk_cw = (const float*)d_in[20];
  const float* blk_cb = (const float*)d_in[21];
  const float* blk_k  = (const float*)d_in[22];
  const float* blk_D  = (const float*)d_in[23];
  const float* blk_ow = (const float*)d_in[24];
  const float* blk_ob = (const float*)d_in[25];
  const float* blk_Wv = (const float*)d_in[26];
  const float* blk_bv = (const float*)d_in[27];
  const float* blk_Wo = (const float*)d_in[28];
  const float* blk_bo = (const float*)d_in[29];
  const float* blk_rw = (const float*)d_in[30];
  const float* blk_rb = (const float*)d_in[31];
  const float* blk_sw = (const float*)d_in[32];
  const float* blk_sb = (const float*)d_in[33];
  const float* fc_w   = (const float*)d_in[34];
  const float* fc_b   = (const float*)d_in[35];
  const float* zc_w   = (const float*)d_in[36];
  const float* zc_b   = (const float*)d_in[37];
  const float* ln_w   = (const float*)d_in[38];
  const float* ln_b   = (const float*)d_in[39];
  float* out = (float*)d_out;

  // ---- workspace layout (floats) ----
  float* ws = (float*)d_ws;
  size_t off = 0;
  auto alloc = [&](size_t n) { float* p = ws + off; off += n; return p; };
  const size_t PL = (size_t)B_ * 25 * L_ * 8;
  const size_t HL = (size_t)B_ * PS_ * L_;
  const size_t RL = (size_t)B_ * RES_ * L_;
  const size_t UL = (size_t)B_ * H2_ * L_;
  float* P0 = alloc(PL);
  float* P1 = alloc(PL);
  float* H  = alloc(HL);
  float* HP = alloc(HL);
  float* CT = alloc((size_t)NF_ * PS_);
  float* ST = alloc((size_t)NF_ * PS_);
  float* SP = alloc((size_t)B_ * NF_ * L_);
  float* NO = alloc(RL);
  float* HX = alloc(RL);
  float* HN = alloc(RL);
  float* SK = alloc(RL);
  float* OU = alloc(RL);
  float* XP = alloc((size_t)B_ * RES_ * LP_);   // padded conv3 input
  float* WC = alloc((size_t)H2_ * 3 * RES_);    // repacked conv3 weights
  float* U  = alloc(UL);
  float* KF = alloc((size_t)S4C_ * H2_ * (FFTN / 2 + 1) * 2);
  float* Y  = alloc((size_t)B_ * S4C_ * H2_ * L_);
  float* Y2 = alloc((size_t)B_ * 2 * H2_ * L_);
  float* HT = alloc(UL);
  float* V  = alloc(UL);
  float* AB = alloc(UL);
  float* TMP = Y2;   // y2 dead after GLU within each block iteration
  float* O1  = U;    // u dead after the loop
  float* O2  = H;    // h dead after ic GEMM

  const int EW = 256;
  auto blocks = [](size_t n) { return (unsigned)((n + 255) / 256); };

  // ---------------- patch embedding ----------------
  k_conv0<<<dim3(L_, B_), 256, 0, stream>>>(x, pi0_w, pi0_b, P0);
  k_gn_gelu<<<B_ * 5, 256, 0, stream>>>(P0, gn0_w, gn0_b);
  k_convw3<<<dim3(L_, B_), 256, 0, stream>>>(P0, pi1_w, pi1_b, P1);
  k_gn_gelu<<<B_ * 5, 256, 0, stream>>>(P1, gn1_w, gn1_b);
  k_convw3<<<dim3(L_, B_), 256, 0, stream>>>(P1, pi2_w, pi2_b, P0);
  k_gn_gelu<<<B_ * 5, 256, 0, stream>>>(P0, gn2_w, gn2_b);
  k_assemble<<<blocks(HL), EW, 0, stream>>>(P0, H);
  k_tables<<<(NF_ * PS_ + 255) / 256, 256, 0, stream>>>(CT, ST);
  k_spec<<<dim3(L_, B_), 128, 0, stream>>>(x, CT, ST, SP);
  // pe += spec_w * spec + spec_b   (ACC GEMM, M=200, K=101 -> KGUARD)
  k_gemm<0, 1, 1><<<gemm_grid(PS_), 128, 0, stream>>>(spec_w, SP, spec_b, H, PS_, NF_, L_, 1.0f);
  k_posconv<<<dim3(PS_, B_), 256, 0, stream>>>(H, pe_w, pe_b, HP);
  // noise = relu(ic_w * h + ic_b)
  k_gemm<1, 0, 0><<<gemm_grid(RES_), 128, 0, stream>>>(ic_w, HP, ic_b, NO, RES_, PS_, L_, 1.0f);
  k_copy<<<blocks(RL), EW, 0, stream>>>(NO, HX, RL);
  k_zero<<<blocks(RL), EW, 0, stream>>>(SK, RL);

  // ---------------- residual blocks ----------------
  for (int i = 0; i < NB_; ++i) {
    k_rms<<<dim3((L_ + 255) / 256, B_), 256, 0, stream>>>(HX, blk_sn + (size_t)i * RES_, HN);
    k_repack_cw<<<(H2_ * 3 * RES_ + 255) / 256, 256, 0, stream>>>(
        blk_cw + (size_t)i * H2_ * RES_ * 3, WC);
    k_xin<<<blocks((size_t)B_ * RES_ * LP_), EW, 0, stream>>>(HX, NO, XP);
    k_conv3<<<gemm_grid(H2_), 128, 0, stream>>>(WC, XP, blk_cb + (size_t)i * H2_, U);
    k_kf<<<dim3(H2_, S4C_), 256, 2 * FFTN * sizeof(float2), stream>>>(
        blk_k + (size_t)i * S4C_ * H2_ * L_, (float2*)KF);
    k_s4<<<dim3(H2_, B_), 256, 3 * FFTN * sizeof(float2), stream>>>(
        U, (const float2*)KF, blk_D + (size_t)i * S4C_ * H2_, Y);
    k_gemm<0, 0, 0><<<gemm_grid(2 * H2_), 128, 0, stream>>>(
        blk_ow + (size_t)i * 2 * H2_ * S4C_ * H2_, Y, blk_ob + (size_t)i * 2 * H2_, Y2,
        2 * H2_, S4C_ * H2_, L_, 1.0f);
    k_glu<<<blocks(UL), EW, 0, stream>>>(Y2, HT);
    k_gemm<0, 0, 0><<<gemm_grid(H2_), 128, 0, stream>>>(
        blk_Wv + (size_t)i * H2_ * H2_, HT, blk_bv + (size_t)i * H2_, V, H2_, H2_, L_, 1.0f);
    k_gemm<0, 0, 0><<<gemm_grid(H2_), 128, 0, stream>>>(
        blk_Wo + (size_t)i * H2_ * H2_, V, blk_bo + (size_t)i * H2_, AB, H2_, H2_, L_, 1.0f);
    k_gate<<<blocks(RL), EW, 0, stream>>>(HT, AB, OU);
    k_gemm<0, 0, 0><<<gemm_grid(RES_), 128, 0, stream>>>(
        blk_rw + (size_t)i * RES_ * RES_, OU, blk_rb + (size_t)i * RES_, TMP, RES_, RES_, L_, 1.0f);
    k_gemm<0, 1, 0><<<gemm_grid(RES_), 128, 0, stream>>>(
        blk_sw + (size_t)i * RES_ * RES_, OU, blk_sb + (size_t)i * RES_, SK, RES_, RES_, L_, 1.0f);
    k_combine<<<blocks(RL), EW, 0, stream>>>(HN, TMP, HX);
  }

  // ---------------- final convs + layernorm ----------------
  const float skscale = 0.40824829046386302f;   // sqrt(1/6), folded into fc GEMM alpha
  k_gemm<1, 0, 0><<<gemm_grid(RES_), 128, 0, stream>>>(fc_w, SK, fc_b, O1, RES_, RES_, L_, skscale);
  k_gemm<0, 0, 0><<<gemm_grid(PS_), 128, 0, stream>>>(zc_w, O1, zc_b, O2, PS_, RES_, L_, 1.0f);
  k_final_ln<<<dim3((L_ + 255) / 256, B_), 256, 0, stream>>>(O2, ln_w, ln_b, out);
}